// TTEmbedding_76871324664463
// MI455X (gfx1250) — compile-verified
//
#include <hip/hip_runtime.h>
#include <hip/hip_bf16.h>

typedef __attribute__((ext_vector_type(2))) float v2f;
typedef __attribute__((ext_vector_type(4))) float v4f;
typedef __attribute__((ext_vector_type(8))) float v8f;

namespace {
constexpr int kM2M3 = 3600;   // M2*M3
constexpr int kM3   = 60;
constexpr int kWaves = 4;     // waves per block (one token per wave)
constexpr int kThreads = kWaves * 32;
constexpr int kEmb = 512;     // N1*N2*N3
// per-wave LDS: t12 [b=8][m=16][s=16] = 2048 f32, emb [512] f32
constexpr int kLdsPerWave = 2048 + 512;
}

// TT-embedding: per token t with id -> (i1,i2,i3):
//   t12[a, b*16+s] = sum_r g1[a,r] * g2[r, b*16+s]      (16x16x4 WMMA, 8 N-tiles x 4 K-chunks)
//   emb[a*8+b, c]  = sum_s t12[a,b,s] * g3[s,c]          (16x16x4 WMMA, 4 M-tiles x 4 K-chunks)
__global__ __launch_bounds__(kThreads)
void tt_embed_wmma_kernel(const long long* __restrict__ x,
                          const float* __restrict__ core1,   // [50][8][16]
                          const float* __restrict__ core2,   // [60][16][8][16]
                          const float* __restrict__ core3,   // [60][16][8]
                          float* __restrict__ out, int T)
{
  __shared__ __align__(16) float lds[kWaves * kLdsPerWave];
  const int lane = threadIdx.x & 31;
  const int wave = threadIdx.x >> 5;
  const int p = blockIdx.x * kWaves + wave;   // token position (s*64+b in memory order)
  if (p >= T) return;

  float* t12s = lds + wave * kLdsPerWave;     // [b][m][s]
  float* embs = t12s + 2048;                  // [(a*8+b)][c]

  const unsigned id = (unsigned)x[p];
  const unsigned i1 = id / kM2M3;
  const unsigned rm = id % kM2M3;
  const unsigned i2 = rm / kM3;
  const unsigned i3 = rm % kM3;

  const int lh  = lane >> 4;   // lane half selects upper K pair in A/B operands
  const int l15 = lane & 15;

  // ---- A operand, contraction 1: g1 as [a(16,pad)][r], K-chunks of 4.
  // A layout: lanes 0-15 hold (M=lane, K=k0..k0+1), lanes 16-31 hold (M=lane-16, K=k0+2..k0+3)
  const float* g1 = core1 + (size_t)i1 * (8 * 16);
  v2f A1[4];
#pragma unroll
  for (int j = 0; j < 4; ++j)
    A1[j] = *(const v2f*)(g1 + (l15 & 7) * 16 + 4 * j + 2 * lh);   // global_load_b64

  // ---- B operand, contraction 2: g3 [s(16)][c(8,pad)], columns clamped to c&7
  const float* g3 = core3 + (size_t)i3 * (16 * 8);
  v2f B3[4];
#pragma unroll
  for (int j = 0; j < 4; ++j) {
    const int s0 = 4 * j + 2 * lh;
    const int c  = l15 & 7;
    B3[j].x = g3[(s0    ) * 8 + c];
    B3[j].y = g3[(s0 + 1) * 8 + c];
  }

  // ---- contraction 1: 8 N-tiles over (b), K=16 over r
  const float* g2 = core2 + (size_t)i2 * (16 * 8 * 16);
#pragma unroll
  for (int bt = 0; bt < 8; ++bt) {
    v8f acc = {0.f,0.f,0.f,0.f,0.f,0.f,0.f,0.f};
#pragma unroll
    for (int j = 0; j < 4; ++j) {
      const int r0 = 4 * j + 2 * lh;
      v2f B2;
      B2.x = g2[(r0    ) * 128 + bt * 16 + l15];
      B2.y = g2[(r0 + 1) * 128 + bt * 16 + l15];
      acc = __builtin_amdgcn_wmma_f32_16x16x4_f32(false, A1[j], false, B2,
                                                  (short)0, acc, false, false);
    }
    // D layout -> LDS [b][m][s]: VGPR v holds (M=v+8*lh, N=l15)
#pragma unroll
    for (int v = 0; v < 8; ++v)
      t12s[bt * 256 + (v + 8 * lh) * 16 + l15] = acc[v];
  }

  // ---- contraction 2: 4 M-tiles over rows (a*8+b), K=16 over s
#pragma unroll
  for (int mt = 0; mt < 4; ++mt) {
    v8f acc = {0.f,0.f,0.f,0.f,0.f,0.f,0.f,0.f};
    const int row = mt * 16 + l15;      // row = a*8+b
    const int a = row >> 3;
    const int b = row & 7;
#pragma unroll
    for (int j = 0; j < 4; ++j) {
      // A chunk from LDS with transpose-by-addressing: lane holds row, K=s in VGPR pair
      v2f At = *(const v2f*)(t12s + b * 256 + a * 16 + 4 * j + 2 * lh); // ds_load_b64
      acc = __builtin_amdgcn_wmma_f32_16x16x4_f32(false, At, false, B3[j],
                                                  (short)0, acc, false, false);
    }
    // keep only valid c<8 columns; rows all valid
#pragma unroll
    for (int v = 0; v < 8; ++v) {
      const int rowD = mt * 16 + v + 8 * lh;
      if (l15 < 8) embs[rowD * 8 + l15] = acc[v];
    }
  }

  // ---- coalesced 2KB row store: 4 x (ds_load_b128 + global_store_b128)
  float* op = out + (size_t)p * kEmb;
#pragma unroll
  for (int rnd = 0; rnd < 4; ++rnd) {
    v4f val = *(const v4f*)(embs + rnd * 128 + lane * 4);
    *(v4f*)(op + rnd * 128 + lane * 4) = val;
  }
}

extern "C" void kernel_launch(void* const* d_in, const int* in_sizes, int n_in,
                              void* d_out, int out_size, void* d_ws, size_t ws_size,
                              hipStream_t stream) {
  const long long* x  = (const long long*)d_in[0];   // int64 token ids [2048,64]
  const float*     c1 = (const float*)d_in[1];       // [50][8][16]
  const float*     c2 = (const float*)d_in[2];       // [60][16][8][16]
  const float*     c3 = (const float*)d_in[3];       // [60][16][8]
  float* out = (float*)d_out;                        // [2048,64,512] f32
  const int T = in_sizes[0];                         // 131072 tokens
  const int blocks = (T + kWaves - 1) / kWaves;
  tt_embed_wmma_kernel<<<blocks, kThreads, 0, stream>>>(x, c1, c2, c3, out, T);
}